// Pool_46935402610871
// MI455X (gfx1250) — compile-verified
//
#include <hip/hip_runtime.h>

#define TOPK 9
#define POOL 20
#define DIM  768
#define NB   32768

typedef __attribute__((ext_vector_type(2))) float v2f;
typedef __attribute__((ext_vector_type(4))) float f4;
typedef __attribute__((ext_vector_type(8))) float v8f;

constexpr int WAVES_PER_BLOCK = 8;
constexpr int ROWS_PER_BLOCK  = WAVES_PER_BLOCK * 16;   // 128

// ---------------- Kernel A: normalize keys[layer] -> wk [20][768] ----------
__global__ __launch_bounds__(256) void normalize_keys(
    const float* __restrict__ keys, const int* __restrict__ layer_p,
    float* __restrict__ wk)
{
  __shared__ float red[256];
  const int row   = blockIdx.x;             // 0..19
  const int layer = layer_p[0];
  const float* src = keys + ((size_t)layer * POOL + row) * DIM;
  float s = 0.f;
  for (int c = threadIdx.x; c < DIM; c += 256) { float v = src[c]; s += v * v; }
  red[threadIdx.x] = s; __syncthreads();
  for (int off = 128; off > 0; off >>= 1) {
    if (threadIdx.x < off) red[threadIdx.x] += red[threadIdx.x + off];
    __syncthreads();
  }
  const float inv = 1.0f / fmaxf(sqrtf(red[0]), 1e-12f);
  for (int c = threadIdx.x; c < DIM; c += 256) wk[row * DIM + c] = src[c] * inv;
}

// ---------------- Kernel B: sim (WMMA f32), top-9, gather-write ------------
__global__ __launch_bounds__(256) void pool_main(
    const float* __restrict__ x, const float* __restrict__ prompts,
    const int* __restrict__ layer_p, const float* __restrict__ wk,
    float* __restrict__ out, float* __restrict__ sim20,
    unsigned int* __restrict__ counts)
{
  __shared__ __align__(16) float lds_kn[POOL * DIM];   // 60 KB normalized keys
  __shared__ __align__(16) float lds_pr[POOL * DIM];   // 60 KB prompts[layer]
  __shared__ float lds_sim[WAVES_PER_BLOCK][16][32];   // 16 KB sim tiles
  __shared__ int   lds_idx[WAVES_PER_BLOCK][16][TOPK]; // 4.5 KB
  __shared__ unsigned int hist[POOL];

  const int tid  = threadIdx.x;
  const int wave = tid >> 5;
  const int lane = tid & 31;
  const int lm   = lane & 15;
  const int half = lane >> 4;
  const int layer = layer_p[0];
  const float* pr = prompts + (size_t)layer * POOL * DIM;   // LEN_PROMPTS == 1

  if (tid < POOL) hist[tid] = 0u;

  // Cooperative stage of keys + prompts into LDS (128-bit ops)
  {
    const f4* kn4 = (const f4*)wk;
    const f4* pr4 = (const f4*)pr;
    f4* lkn4 = (f4*)lds_kn;
    f4* lpr4 = (f4*)lds_pr;
    const int n4 = POOL * DIM / 4;                    // 3840
    for (int i = tid; i < n4; i += 256) { lkn4[i] = kn4[i]; lpr4[i] = pr4[i]; }
  }
  __syncthreads();

  const int r0 = blockIdx.x * ROWS_PER_BLOCK + wave * 16;

  // A fragment: lane L holds row (r0 + L&15), columns k + 2*(L>>4) + {0,1}
  const float* aptr  = x + (size_t)(r0 + lm) * DIM + 2 * half;
  // B fragments from LDS keys: N-tile0 pools 0..15, N-tile1 pools 16..19
  const float* b0ptr = lds_kn + lm * DIM + 2 * half;
  const bool   v1    = (lm < 4);
  const float* b1ptr = lds_kn + (v1 ? (16 + lm) : 0) * DIM + 2 * half;

  v8f acc0 = {}; v8f acc1 = {};
  float ns = 0.f;

#pragma unroll 2
  for (int k = 0; k < DIM; k += 4) {
    v2f a  = *(const v2f*)(aptr  + k);
    v2f b0 = *(const v2f*)(b0ptr + k);
    v2f bt = *(const v2f*)(b1ptr + k);
    v2f b1; b1.x = v1 ? bt.x : 0.f; b1.y = v1 ? bt.y : 0.f;
    ns += a.x * a.x + a.y * a.y;
    acc0 = __builtin_amdgcn_wmma_f32_16x16x4_f32(false, a, false, b0,
                                                 (short)0, acc0, false, false);
    acc1 = __builtin_amdgcn_wmma_f32_16x16x4_f32(false, a, false, b1,
                                                 (short)0, acc1, false, false);
  }

  // Full row |x|^2: lanes L and L+16 cover complementary columns of same row
  const float nsTot = ns + __shfl_xor(ns, 16, 32);

  // Spill sim tile to LDS: VGPR v of C/D holds M = v + 8*half, N = lm (+16)
#pragma unroll
  for (int v = 0; v < 8; v++) {
    lds_sim[wave][v + 8 * half][lm]      = acc0[v];
    lds_sim[wave][v + 8 * half][16 + lm] = acc1[v];
  }
  __syncthreads();

  // Lanes 0..15: normalize row, record sim20, select top-9 (stable desc)
  if (lane < 16) {
    const float inv = 1.0f / fmaxf(sqrtf(nsTot), 1e-12f);
    float vals[POOL];
#pragma unroll
    for (int j = 0; j < POOL; j++) vals[j] = lds_sim[wave][lane][j] * inv;
    const int b = r0 + lane;
    if (b < POOL) {
      for (int j = 0; j < POOL; j++) sim20[b * POOL + j] = vals[j];
    }
    for (int s = 0; s < TOPK; s++) {
      float best = -INFINITY; int bi = 0;
#pragma unroll
      for (int j = 0; j < POOL; j++) {
        if (vals[j] > best) { best = vals[j]; bi = j; }
      }
      lds_idx[wave][lane][s] = bi;
      vals[bi] = -INFINITY;
      atomicAdd(&hist[bi], 1u);
    }
  }
  __syncthreads();

  // Gather-write: 16 rows * 9 prompts * 3 KB, coalesced NT 128-bit stores
  {
    const f4* lp4  = (const f4*)lds_pr;
    f4*       out4 = (f4*)out;
    for (int cm = 0; cm < 16 * TOPK; cm++) {
      const int m = cm / TOPK, s = cm - m * TOPK;
      const int idx = lds_idx[wave][m][s];
      const f4* src = lp4 + idx * (DIM / 4);
      f4* dst = out4 + ((size_t)(r0 + m) * TOPK + s) * (DIM / 4);
#pragma unroll
      for (int u = 0; u < 6; u++)
        __builtin_nontemporal_store(src[lane + 32 * u], dst + lane + 32 * u);
    }
  }

  if (tid < POOL) atomicAdd(&counts[tid], hist[tid]);
}

// ---------------- Kernel C: dist = 1 - sum(count[i]*rowsum(sim20[i]))/N ----
__global__ void finalize(const float* __restrict__ sim20,
                         const unsigned int* __restrict__ counts,
                         float* __restrict__ dist_out)
{
  __shared__ float S[POOL];
  const int t = threadIdx.x;
  if (t < POOL) {
    float s = 0.f;
    for (int j = 0; j < POOL; j++) s += sim20[t * POOL + j];
    S[t] = s;
  }
  __syncthreads();
  if (t == 0) {
    double tot = 0.0;
    for (int i = 0; i < POOL; i++) tot += (double)counts[i] * (double)S[i];
    const double denom = (double)NB * TOPK * POOL;   // 5,898,240
    *dist_out = (float)(1.0 - tot / denom);
  }
}

extern "C" void kernel_launch(void* const* d_in, const int* in_sizes, int n_in,
                              void* d_out, int out_size, void* d_ws, size_t ws_size,
                              hipStream_t stream) {
  const float* x       = (const float*)d_in[0];
  const float* keys    = (const float*)d_in[1];
  const float* prompts = (const float*)d_in[2];
  const int*   layer   = (const int*)d_in[3];
  float* out = (float*)d_out;

  float* wk            = (float*)d_ws;              // 20*768 normalized keys
  float* sim20         = wk + POOL * DIM;           // 20*20
  unsigned int* counts = (unsigned int*)(sim20 + POOL * POOL);  // 20

  hipMemsetAsync(counts, 0, POOL * sizeof(unsigned int), stream);
  normalize_keys<<<POOL, 256, 0, stream>>>(keys, layer, wk);
  pool_main<<<NB / ROWS_PER_BLOCK, 256, 0, stream>>>(x, prompts, layer, wk, out,
                                                     sim20, counts);
  finalize<<<1, 32, 0, stream>>>(sim20, counts, out + (size_t)NB * TOPK * DIM);
}